// RobustEnhancedSpectralAttention_21268678049967
// MI455X (gfx1250) — compile-verified
//
#include <hip/hip_runtime.h>
#include <hip/hip_bf16.h>

typedef __attribute__((ext_vector_type(2))) float v2f;
typedef __attribute__((ext_vector_type(8))) float v8f;

// Problem constants (fixed by the reference)
static constexpr int BB = 4, CC = 256, HH = 256, WW = 256;
static constexpr int RD = 93, TRD = 279, NB = 31;
static constexpr int MP = 288;   // qkv rows padded (18 tiles)
static constexpr int KP = 96;    // fusion K padded (24 wmma steps)

// Workspace layout (in floats)
static constexpr long XP4_OFF   = 0;        // 4*256*16   = 16384
static constexpr long XP8_OFF   = 16384;    // 4*256*64   = 65536
static constexpr long XP16_OFF  = 81920;    // 4*256*256  = 262144
static constexpr long QKV4_OFF  = 344064;   // 4*288*16   = 18432
static constexpr long QKV8_OFF  = 362496;   // 4*288*64   = 73728
static constexpr long QKV16_OFF = 436224;   // 4*288*256  = 294912
static constexpr long OUT4_OFF  = 731136;   // 4*96*16    = 6144
static constexpr long OUT8_OFF  = 737280;   // 4*96*64    = 24576
static constexpr long OUT16_OFF = 761856;   // 4*96*256   = 98304
static constexpr long FO4_OFF   = 860160;   // 4*256*16   = 16384
static constexpr long FO8_OFF   = 876544;   // 4*256*64   = 65536
static constexpr long FO16_OFF  = 942080;   // 4*256*256  = 262144
static constexpr long CONVP_OFF = 1204224;  // 3*288*256  = 221184
static constexpr long FUSP_OFF  = 1425408;  // 3*256*96   = 73728
static constexpr long PART_OFF  = 1499136;  // 4*256*2    = 2048
static constexpr long STAT_OFF  = 1501184;  // 4*32*2     = 256

// ---------------------------------------------------------------------------
// K0: zero-pad weights so all GEMMs are guard-free.
// ---------------------------------------------------------------------------
__global__ __launch_bounds__(256) void pad_conv_kernel(
    const float* __restrict__ w0, const float* __restrict__ w1,
    const float* __restrict__ w2, float* __restrict__ convp)
{
    int r = blockIdx.x, si = blockIdx.y, t = threadIdx.x;
    const float* src = si == 0 ? w0 : (si == 1 ? w1 : w2);
    convp[((long)si * MP + r) * CC + t] = (r < TRD) ? src[(long)r * CC + t] : 0.f;
}

__global__ __launch_bounds__(96) void pad_fus_kernel(
    const float* __restrict__ fw, float* __restrict__ fuswp)
{
    int r = blockIdx.x, si = blockIdx.y, t = threadIdx.x;
    int off = si * RD;
    fuswp[((long)si * CC + r) * KP + t] = (t < RD) ? fw[(long)r * TRD + off + t] : 0.f;
}

// ---------------------------------------------------------------------------
// K1: average-pool x (B,C,256,256) -> xp16/xp8/xp4.  One block per (b,c).
// xp8/xp4 are exact 2x2 re-poolings of xp16 (avg of avg).
// ---------------------------------------------------------------------------
__global__ __launch_bounds__(256) void pool_kernel(
    const float* __restrict__ x, float* __restrict__ xp16,
    float* __restrict__ xp8, float* __restrict__ xp4)
{
    __shared__ float sm16[256];
    __shared__ float sm8[64];
    int bc = blockIdx.x;
    int t  = threadIdx.x;
    const float* img = x + (long)bc * (HH * WW);
    int ti = t >> 4, tj = t & 15;
    const float* p = img + (long)ti * 16 * WW + tj * 16;
    float s = 0.f;
    for (int r = 0; r < 16; ++r) {
        const float4* q = (const float4*)(p + (long)r * WW);
        #pragma unroll
        for (int c4 = 0; c4 < 4; ++c4) {
            float4 v = q[c4];
            s += v.x + v.y + v.z + v.w;
        }
    }
    s *= (1.f / 256.f);
    xp16[(long)bc * 256 + t] = s;
    sm16[t] = s;
    __syncthreads();
    if (t < 64) {
        int i = t >> 3, j = t & 7;
        float v = 0.25f * (sm16[(2*i)*16 + 2*j]   + sm16[(2*i)*16 + 2*j+1] +
                           sm16[(2*i+1)*16 + 2*j] + sm16[(2*i+1)*16 + 2*j+1]);
        xp8[(long)bc * 64 + t] = v;
        sm8[t] = v;
    }
    __syncthreads();
    if (t < 16) {
        int i = t >> 2, j = t & 3;
        float v = 0.25f * (sm8[(2*i)*8 + 2*j]   + sm8[(2*i)*8 + 2*j+1] +
                           sm8[(2*i+1)*8 + 2*j] + sm8[(2*i+1)*8 + 2*j+1]);
        xp4[(long)bc * 16 + t] = v;
    }
}

// ---------------------------------------------------------------------------
// Guard-free f32 WMMA GEMM.  C[b] = A * B[b], one wave per 16x16 C tile.
// A row-major MPAD x K (K = KSTEPS*4, zero-padded), B row-major K x N,
// C row-major MPAD x N.  Static K and N -> unrolled by 8, immediate-offset
// loads; A fragment is one aligned b64 load per step.
// Fragment layouts (ISA 7.12.2, wave32):
//   A (16x4): lane l, vgpr v: M = l&15, K = (l>>4)*2 + v
//   B (4x16): lane l, vgpr v: N = l&15, K = (l>>4)*2 + v
//   C:        lane l, vgpr r: M = (l>>4)*8 + r, N = l&15
// ---------------------------------------------------------------------------
template <int KSTEPS, int N>
__global__ __launch_bounds__(32) void wmma_gemm_kernel(
    float* __restrict__ Cm, const float* __restrict__ Am,
    const float* __restrict__ Bm, long c_bstride, long b_bstride)
{
    constexpr int K = KSTEPS * 4;
    constexpr int NT = N / 16;
    int mtile = blockIdx.x / NT;
    int ntile = blockIdx.x - mtile * NT;
    const float* Bb = Bm + (long)blockIdx.y * b_bstride;
    float*       Cb = Cm + (long)blockIdx.y * c_bstride;
    int l    = threadIdx.x;
    int half = l >> 4;
    int lo   = l & 15;
    const float* Arow = Am + (long)(mtile * 16 + lo) * K + half * 2;
    const float* Bcol = Bb + (long)(half * 2) * N + ntile * 16 + lo;
    v8f acc = {0.f, 0.f, 0.f, 0.f, 0.f, 0.f, 0.f, 0.f};
    #pragma unroll 8
    for (int ks = 0; ks < KSTEPS; ++ks) {
        v2f a = *(const v2f*)(Arow + ks * 4);
        v2f b;
        b.x = Bcol[(long)ks * 4 * N];
        b.y = Bcol[(long)(ks * 4 + 1) * N];
        acc = __builtin_amdgcn_wmma_f32_16x16x4_f32(
            false, a, false, b, (short)0, acc, false, false);
    }
    float* Crow = Cb + (long)(mtile * 16 + half * 8) * N + ntile * 16 + lo;
    #pragma unroll
    for (int r = 0; r < 8; ++r) Crow[(long)r * N] = acc[r];
}

// ---------------------------------------------------------------------------
// K2b: tiny spectral attention.  One thread per (b, pixel) of the s*s grid.
// qkv layout: [b][288][S] (rows >=279 unused); out layout: [b][96][S]
// (rows 93..95 written as zero so the fusion GEMM needs no K guard).
// ---------------------------------------------------------------------------
__global__ void attn_kernel(const float* __restrict__ qkv,
                            const float* __restrict__ sw,
                            float* __restrict__ outp, int S)
{
    int b = blockIdx.x;
    int pix = threadIdx.x;
    if (pix >= S) return;
    const float* qb = qkv + (long)b * MP * S;
    float* ob = outp + (long)b * KP * S;
    const float sc = 0.57735026918962576f;  // 3^-0.5
    for (int g = 0; g < 3; ++g) {
        float qreg[NB];
        #pragma unroll
        for (int n = 0; n < NB; ++n) qreg[n] = qb[(long)(g * NB + n) * S + pix];
        float qsw[NB];
        #pragma unroll
        for (int m = 0; m < NB; ++m) {
            float acc = 0.f;
            #pragma unroll
            for (int n = 0; n < NB; ++n) acc += qreg[n] * sw[n * NB + m];
            qsw[m] = acc;
        }
        float a0 = 0.f, a1 = 0.f, a2 = 0.f;
        #pragma unroll
        for (int m = 0; m < NB; ++m) {
            a0 += qsw[m] * qb[(long)(RD + 0 * NB + m) * S + pix];
            a1 += qsw[m] * qb[(long)(RD + 1 * NB + m) * S + pix];
            a2 += qsw[m] * qb[(long)(RD + 2 * NB + m) * S + pix];
        }
        a0 *= sc; a1 *= sc; a2 *= sc;
        float mx = fmaxf(a0, fmaxf(a1, a2));
        float e0 = expf(a0 - mx), e1 = expf(a1 - mx), e2 = expf(a2 - mx);
        float inv = 1.f / (e0 + e1 + e2);
        float w0 = e0 * inv, w1 = e1 * inv, w2 = e2 * inv;
        #pragma unroll
        for (int n = 0; n < NB; ++n) {
            float v0 = qb[(long)(2 * RD + 0 * NB + n) * S + pix];
            float v1 = qb[(long)(2 * RD + 1 * NB + n) * S + pix];
            float v2 = qb[(long)(2 * RD + 2 * NB + n) * S + pix];
            ob[(long)(g * NB + n) * S + pix] = w0 * v0 + w1 * v1 + w2 * v2;
        }
    }
    // zero K-pad rows for the fusion GEMM
    ob[(long)93 * S + pix] = 0.f;
    ob[(long)94 * S + pix] = 0.f;
    ob[(long)95 * S + pix] = 0.f;
}

// ---------------------------------------------------------------------------
// K3/K5: streaming pass.  One block per (b,c); builds U_s = fo_s * Mw^T in
// LDS, then walks rows with incremental piecewise-linear interpolation in h
// (all knots are multiples of 8).  MODE 0: accumulate sum / sumsq partials.
// MODE 1: write normalized output.
// ---------------------------------------------------------------------------
__device__ __forceinline__ void seg_params(float raw, float fmax,
                                           int& L, int& Hh, float& Wf, bool& lin)
{
    float s = fminf(fmaxf(raw, 0.f), fmax);
    L = (int)s;
    Hh = (float)(L + 1) <= fmax ? L + 1 : L;
    Wf = s - (float)L;
    lin = (raw >= 0.f && raw < fmax);
}

template <int MODE>
__global__ __launch_bounds__(256) void apply_kernel(
    const float* __restrict__ x,
    const float* __restrict__ fo4, const float* __restrict__ fo8,
    const float* __restrict__ fo16,
    const float* __restrict__ fusion_b, const float* __restrict__ gn_w,
    const float* __restrict__ gn_b, const float* __restrict__ stats,
    float* __restrict__ partials, float* __restrict__ outp)
{
    __shared__ float U16[16 * 256];
    __shared__ float U8[8 * 256];
    __shared__ float U4[4 * 256];
    __shared__ float f16s[256];
    __shared__ float f8s[64];
    __shared__ float f4s[16];
    __shared__ float red[256];

    int bc = blockIdx.x;
    int b = bc >> 8, c = bc & 255;
    int t = threadIdx.x;

    f16s[t] = fo16[(long)bc * 256 + t];
    if (t < 64) f8s[t] = fo8[(long)bc * 64 + t];
    if (t < 16) f4s[t] = fo4[(long)bc * 16 + t];
    __syncthreads();

    // Row-interp tables along w for this thread's column t.
    {
        float wp = (float)t + 0.5f;
        int L, Hh; float Wf; bool lin;
        seg_params(wp * 0.0625f - 0.5f, 15.f, L, Hh, Wf, lin);
        #pragma unroll
        for (int i = 0; i < 16; ++i) {
            float a = f16s[i * 16 + L], bb = f16s[i * 16 + Hh];
            U16[i * 256 + t] = a + Wf * (bb - a);
        }
        seg_params(wp * 0.03125f - 0.5f, 7.f, L, Hh, Wf, lin);
        #pragma unroll
        for (int i = 0; i < 8; ++i) {
            float a = f8s[i * 8 + L], bb = f8s[i * 8 + Hh];
            U8[i * 256 + t] = a + Wf * (bb - a);
        }
        seg_params(wp * 0.015625f - 0.5f, 3.f, L, Hh, Wf, lin);
        #pragma unroll
        for (int i = 0; i < 4; ++i) {
            float a = f4s[i * 4 + L], bb = f4s[i * 4 + Hh];
            U4[i * 256 + t] = a + Wf * (bb - a);
        }
    }
    __syncthreads();

    float fb = fusion_b[c];
    float scl = 0.f, off = 0.f;
    if (MODE == 1) {
        int g = c >> 3;
        float mean = stats[(b * 32 + g) * 2];
        float inv  = stats[(b * 32 + g) * 2 + 1];
        scl = inv * gn_w[c];
        off = gn_b[c] - mean * scl;
    }

    const float* xim = x + (long)bc * (HH * WW);
    float* oim = outp + (long)bc * (HH * WW);
    float s1 = 0.f, s2 = 0.f;

    for (int h0 = 0; h0 < 256; h0 += 8) {
        float hp = (float)h0 + 0.5f;
        float u = fb, du = 0.f;
        {
            int L, Hh; float Wf; bool lin;
            float raw = hp * 0.0625f - 0.5f;
            seg_params(raw, 15.f, L, Hh, Wf, lin);
            float a = U16[L * 256 + t], bb = U16[Hh * 256 + t];
            u += a + Wf * (bb - a);
            if (lin) du += 0.0625f * (bb - a);
        }
        {
            int L, Hh; float Wf; bool lin;
            float raw = hp * 0.03125f - 0.5f;
            seg_params(raw, 7.f, L, Hh, Wf, lin);
            float a = U8[L * 256 + t], bb = U8[Hh * 256 + t];
            u += a + Wf * (bb - a);
            if (lin) du += 0.03125f * (bb - a);
        }
        {
            int L, Hh; float Wf; bool lin;
            float raw = hp * 0.015625f - 0.5f;
            seg_params(raw, 3.f, L, Hh, Wf, lin);
            float a = U4[L * 256 + t], bb = U4[Hh * 256 + t];
            u += a + Wf * (bb - a);
            if (lin) du += 0.015625f * (bb - a);
        }
        const float* xr = xim + (long)h0 * 256 + t;
        if (MODE == 1) {
            float* orow = oim + (long)h0 * 256 + t;
            #pragma unroll
            for (int hh = 0; hh < 8; ++hh) {
                float v = xr[hh * 256] + u;
                orow[hh * 256] = v * scl + off;
                u += du;
            }
        } else {
            #pragma unroll
            for (int hh = 0; hh < 8; ++hh) {
                float v = xr[hh * 256] + u;
                s1 += v;
                s2 = fmaf(v, v, s2);
                u += du;
            }
        }
    }

    if (MODE == 0) {
        red[t] = s1;
        __syncthreads();
        for (int st = 128; st > 0; st >>= 1) {
            if (t < st) red[t] += red[t + st];
            __syncthreads();
        }
        float tot1 = red[0];
        __syncthreads();
        red[t] = s2;
        __syncthreads();
        for (int st = 128; st > 0; st >>= 1) {
            if (t < st) red[t] += red[t + st];
            __syncthreads();
        }
        if (t == 0) {
            partials[bc * 2]     = tot1;
            partials[bc * 2 + 1] = red[0];
        }
    }
}

// ---------------------------------------------------------------------------
// K4: finalize GroupNorm stats.  One thread per (b, group).
// ---------------------------------------------------------------------------
__global__ void stats_kernel(const float* __restrict__ partials,
                             float* __restrict__ stats)
{
    int i = threadIdx.x;
    if (i >= 128) return;
    int b = i >> 5, g = i & 31;
    float s1 = 0.f, s2 = 0.f;
    #pragma unroll
    for (int k = 0; k < 8; ++k) {
        int bc = b * 256 + g * 8 + k;
        s1 += partials[bc * 2];
        s2 += partials[bc * 2 + 1];
    }
    const float cnt = 8.f * 65536.f;
    float mean = s1 / cnt;
    float var  = s2 / cnt - mean * mean;
    stats[i * 2]     = mean;
    stats[i * 2 + 1] = rsqrtf(var + 1e-5f);
}

// ---------------------------------------------------------------------------
extern "C" void kernel_launch(void* const* d_in, const int* in_sizes, int n_in,
                              void* d_out, int out_size, void* d_ws, size_t ws_size,
                              hipStream_t stream)
{
    (void)in_sizes; (void)n_in; (void)out_size; (void)ws_size;
    const float* x        = (const float*)d_in[0];
    const float* cw0      = (const float*)d_in[1];
    const float* cw1      = (const float*)d_in[2];
    const float* cw2      = (const float*)d_in[3];
    const float* sw[3]    = {(const float*)d_in[4], (const float*)d_in[5], (const float*)d_in[6]};
    const float* fusion_w = (const float*)d_in[7];
    const float* fusion_b = (const float*)d_in[8];
    const float* gn_w     = (const float*)d_in[9];
    const float* gn_b     = (const float*)d_in[10];
    float* out = (float*)d_out;
    float* ws  = (float*)d_ws;

    float* xp[3]   = {ws + XP4_OFF,  ws + XP8_OFF,  ws + XP16_OFF};
    float* qkv[3]  = {ws + QKV4_OFF, ws + QKV8_OFF, ws + QKV16_OFF};
    float* osm[3]  = {ws + OUT4_OFF, ws + OUT8_OFF, ws + OUT16_OFF};
    float* fo[3]   = {ws + FO4_OFF,  ws + FO8_OFF,  ws + FO16_OFF};
    float* convp   = ws + CONVP_OFF;   // 3 x [288 x 256]
    float* fuswp   = ws + FUSP_OFF;    // 3 x [256 x 96]
    float* part    = ws + PART_OFF;
    float* stats   = ws + STAT_OFF;

    // K0: zero-padded weight staging (guard-free GEMMs)
    pad_conv_kernel<<<dim3(MP, 3), dim3(CC), 0, stream>>>(cw0, cw1, cw2, convp);
    pad_fus_kernel<<<dim3(CC, 3), dim3(KP), 0, stream>>>(fusion_w, fuswp);

    // K1: pooling (reads x once)
    pool_kernel<<<dim3(BB * CC), dim3(256), 0, stream>>>(x, xp[2], xp[1], xp[0]);

    // K2: per-scale small compute (WMMA GEMM -> attention -> WMMA GEMM)
    // scale 4 (S=16)
    wmma_gemm_kernel<64, 16><<<dim3(18 * 1, BB), dim3(32), 0, stream>>>(
        qkv[0], convp + 0L * MP * CC, xp[0], (long)MP * 16, (long)CC * 16);
    attn_kernel<<<dim3(BB), dim3(16), 0, stream>>>(qkv[0], sw[0], osm[0], 16);
    wmma_gemm_kernel<24, 16><<<dim3(16 * 1, BB), dim3(32), 0, stream>>>(
        fo[0], fuswp + 0L * CC * KP, osm[0], (long)CC * 16, (long)KP * 16);
    // scale 8 (S=64)
    wmma_gemm_kernel<64, 64><<<dim3(18 * 4, BB), dim3(32), 0, stream>>>(
        qkv[1], convp + 1L * MP * CC, xp[1], (long)MP * 64, (long)CC * 64);
    attn_kernel<<<dim3(BB), dim3(64), 0, stream>>>(qkv[1], sw[1], osm[1], 64);
    wmma_gemm_kernel<24, 64><<<dim3(16 * 4, BB), dim3(32), 0, stream>>>(
        fo[1], fuswp + 1L * CC * KP, osm[1], (long)CC * 64, (long)KP * 64);
    // scale 16 (S=256)
    wmma_gemm_kernel<64, 256><<<dim3(18 * 16, BB), dim3(32), 0, stream>>>(
        qkv[2], convp + 2L * MP * CC, xp[2], (long)MP * 256, (long)CC * 256);
    attn_kernel<<<dim3(BB), dim3(256), 0, stream>>>(qkv[2], sw[2], osm[2], 256);
    wmma_gemm_kernel<24, 256><<<dim3(16 * 16, BB), dim3(32), 0, stream>>>(
        fo[2], fuswp + 2L * CC * KP, osm[2], (long)CC * 256, (long)KP * 256);

    // K3: stats pass over x + low-rank field
    apply_kernel<0><<<dim3(BB * CC), dim3(256), 0, stream>>>(
        x, fo[0], fo[1], fo[2], fusion_b, gn_w, gn_b, stats, part, out);
    // K4: finalize per-(b,group) mean / inv-std
    stats_kernel<<<dim3(1), dim3(128), 0, stream>>>(part, stats);
    // K5: normalize + write output
    apply_kernel<1><<<dim3(BB * CC), dim3(256), 0, stream>>>(
        x, fo[0], fo[1], fo[2], fusion_b, gn_w, gn_b, stats, part, out);
}